// MyLIFNode_15083925144245
// MI455X (gfx1250) — compile-verified
//
#include <hip/hip_runtime.h>

// LIF forward: spikes over T=8 steps of a [8,32,128,32,32] fp32 tensor.
// Memory-bound streaming recurrence: 128 MiB in + 128 MiB out = 256 MiB
// -> ~11.5us floor @ 23.3 TB/s HBM; compute (~134 MFLOP) is negligible.
// Strategy: 1 thread = 1 float4 spatial slot; membrane kept in registers;
// all 8 timestep tiles fetched up-front with CDNA5 async global->LDS b128
// transfers (ASYNCcnt) using TH_LOAD_NT (x is read exactly once -> don't
// pollute the 192MB L2), consumed with graduated s_wait_asynccnt; spikes
// written with non-temporal b128 stores (never re-read).

#define TPB    256
#define TSTEPS 8

typedef __attribute__((ext_vector_type(4))) float v4f;
typedef __attribute__((address_space(3))) void lds_void;

#define S_WAIT_ASYNCCNT(n) asm volatile("s_wait_asynccnt " #n ::: "memory")

__global__ __launch_bounds__(TPB) void lif_fwd_async(const float* __restrict__ x,
                                                     float* __restrict__ out,
                                                     int n4) {
    __shared__ v4f tile[TSTEPS][TPB];

    const int tid = threadIdx.x;
    const long long gidx = (long long)blockIdx.x * TPB + tid;
    const long long stride4 = (long long)n4;   // float4 units per timestep
    if (gidx >= n4) return;

    // Issue all 8 timestep loads asynchronously into LDS (ASYNCcnt += 8).
    // Each lane stages its own 16B slot, so consumption needs no barrier,
    // only the in-order ASYNCcnt waits. NT hint: x is single-use streaming.
    const v4f* xg = (const v4f*)x + gidx;
#pragma unroll
    for (int t = 0; t < TSTEPS; ++t) {
        unsigned long long ga =
            (unsigned long long)(const void*)(xg + (long long)t * stride4);
        unsigned la =
            (unsigned)(unsigned long long)(lds_void*)&tile[t][tid];
        asm volatile("global_load_async_to_lds_b128 %0, %1, off th:TH_LOAD_NT"
                     :: "v"(la), "v"(ga)
                     : "memory");
    }

    v4f m = {0.0f, 0.0f, 0.0f, 0.0f};   // v_reset = 0
    v4f* og = (v4f*)out + gidx;
    const float tau = 0.5f;
    const float th  = 0.5f;

#define LIF_STEP(T_, CNT_)                                                  \
    {                                                                       \
        S_WAIT_ASYNCCNT(CNT_);             /* tile T_ resident in LDS */    \
        v4f xv = tile[T_][tid];                                             \
        m = m + (xv - m) * tau;            /* leaky integrate */            \
        v4f s;                                                              \
        s.x = (m.x - th > 0.0f) ? 1.0f : 0.0f;                              \
        s.y = (m.y - th > 0.0f) ? 1.0f : 0.0f;                              \
        s.z = (m.z - th > 0.0f) ? 1.0f : 0.0f;                              \
        s.w = (m.w - th > 0.0f) ? 1.0f : 0.0f;                              \
        m.x = (s.x != 0.0f) ? 0.0f : m.x;  /* reset-to-zero on spike */     \
        m.y = (s.y != 0.0f) ? 0.0f : m.y;                                   \
        m.z = (s.z != 0.0f) ? 0.0f : m.z;                                   \
        m.w = (s.w != 0.0f) ? 0.0f : m.w;                                   \
        __builtin_nontemporal_store(s, og + (long long)T_ * stride4);       \
    }

    LIF_STEP(0, 7)
    LIF_STEP(1, 6)
    LIF_STEP(2, 5)
    LIF_STEP(3, 4)
    LIF_STEP(4, 3)
    LIF_STEP(5, 2)
    LIF_STEP(6, 1)
    LIF_STEP(7, 0)
#undef LIF_STEP
}

extern "C" void kernel_launch(void* const* d_in, const int* in_sizes, int n_in,
                              void* d_out, int out_size, void* d_ws, size_t ws_size,
                              hipStream_t stream) {
    const float* x = (const float*)d_in[0];
    float* out = (float*)d_out;

    const int total = in_sizes[0];              // T*B*C*H*W = 33,554,432
    const int n_per_t = total / TSTEPS;         // 4,194,304 floats / step
    const int n4 = n_per_t / 4;                 // 1,048,576 float4 slots

    const int blocks = (n4 + TPB - 1) / TPB;    // 4096
    lif_fwd_async<<<blocks, TPB, 0, stream>>>(x, out, n4);
}